// GATdgl_49460843381581
// MI455X (gfx1250) — compile-verified
//
#include <hip/hip_runtime.h>
#include <math.h>

// ---------------------------------------------------------------------------
// GAT forward on gfx1250 (MI455X).
//   * Dense GEMMs: V_WMMA_F32_16X16X4_F32 (exact fp32), 8-wave 64x64 tiles,
//     operands staged through LDS with GLOBAL_LOAD_ASYNC_TO_LDS_B128
//     (ASYNCcnt / s_wait_asynccnt) -- the CDNA5 async data-mover path.
//   * Edge segment softmax/aggregation: encoded atomic-max + atomic f32 adds.
// ---------------------------------------------------------------------------

#define NN     10000
#define EE     160000
#define FD     128      // F == D == 128
#define HH     4
#define HD     512      // H*D
#define BB     64
#define CC     2
#define FFF    768
#define SLOPE  0.2f

typedef float v2f __attribute__((ext_vector_type(2)));
typedef float v8f __attribute__((ext_vector_type(8)));

__device__ __forceinline__ float leaky(float x) { return x >= 0.f ? x : SLOPE * x; }

// order-preserving float <-> u32 encoding (for atomicMax on floats, any sign)
__device__ __forceinline__ unsigned encf(float f) {
    unsigned u = __float_as_uint(f);
    return (u & 0x80000000u) ? ~u : (u | 0x80000000u);
}
__device__ __forceinline__ float decf(unsigned u) {
    return (u & 0x80000000u) ? __uint_as_float(u & 0x7fffffffu) : __uint_as_float(~u);
}
#define ENC_NEG_INF 0x007FFFFFu   // encf(-inf)

// ---------------------------------------------------------------------------
// Tiled fp32 WMMA GEMM: C[M,N] = A[M,K] @ B[K,N] (+bias[N] if given).
//   block = 256 threads (8 waves), block tile 64x64, K staged by 64 via
//   async global->LDS copies. Wave w computes subtiles (m=w>>1, n=2(w&1)+{0,1}).
// LDS strides: SA=68 (=4 mod 64) -> A b64 fragment reads are bank-conflict
// free; SB=80 (2*SB=32 mod 64) -> the two lane-halves (rows k / k+2) use
// disjoint bank ranges.
// ---------------------------------------------------------------------------
#define BM 64
#define BN 64
#define BK 64
#define SA 68
#define SB 80
#define GEMM_LDS_BYTES ((BM * SA + BK * SB) * 4)   // 37888 B

__global__ __launch_bounds__(256)
void wmma_gemm_tiled(const float* __restrict__ A, const float* __restrict__ B,
                     const float* __restrict__ bias, float* __restrict__ C,
                     int M, int N, int K) {
    extern __shared__ float lds[];
    float* Als = lds;                       // [BM][SA]
    float* Bls = lds + BM * SA;             // [BK][SB]
    const unsigned BlsByteOff = (unsigned)(BM * SA * 4);

    const int tid  = threadIdx.x;
    const int wid  = tid >> 5;
    const int lane = tid & 31;
    const int half = lane >> 4;             // 0 or 1 (K-half of the fragment)
    const int ll   = lane & 15;

    const int mi    = wid >> 1;             // 0..3   row subtile
    const int ni0   = (wid & 1) * 2;        // 0 or 2 first col subtile
    const int rowL  = mi * 16 + ll;         // local A row     (0..63)
    const int colL0 = ni0 * 16 + ll;        // local B col #0
    const int colL1 = colL0 + 16;           // local B col #1

    const int rowBase = blockIdx.x * BM;
    const int colBase = blockIdx.y * BN;

    v8f acc0 = {}, acc1 = {};

    for (int kb = 0; kb < K; kb += BK) {
        __syncthreads();                    // previous compute done reading LDS

        // ---- stage A tile: BM x BK  (1024 x 16B chunks, 4 per thread) ----
        for (int c = tid; c < BM * (BK / 4); c += 256) {
            int m = c >> 4, q = c & 15;
            int gr = rowBase + m; if (gr > M - 1) gr = M - 1;   // clamp tail
            const float* ga = A + (size_t)gr * K + kb + q * 4;
            unsigned lo = (unsigned)((m * SA + q * 4) * 4);
            asm volatile("global_load_async_to_lds_b128 %0, %1, off"
                         :: "v"(lo), "v"(ga) : "memory");
        }
        // ---- stage B tile: BK x BN ----
        for (int c = tid; c < BK * (BN / 4); c += 256) {
            int kk = c >> 4, q = c & 15;
            const float* gb = B + (size_t)(kb + kk) * N + colBase + q * 4;
            unsigned lo = BlsByteOff + (unsigned)((kk * SB + q * 4) * 4);
            asm volatile("global_load_async_to_lds_b128 %0, %1, off"
                         :: "v"(lo), "v"(gb) : "memory");
        }
        asm volatile("s_wait_asynccnt 0x0" ::: "memory");
        __syncthreads();

        // ---- compute: 16 K-steps, 2 WMMAs each (A fragment reused) ----
#pragma unroll
        for (int k = 0; k < BK; k += 4) {
            const int ka = k + (half << 1);
            v2f a = *(const v2f*)(Als + rowL * SA + ka);        // ds_load_b64
            v2f b0; b0.x = Bls[ka * SB + colL0];
                    b0.y = Bls[(ka + 1) * SB + colL0];
            v2f b1; b1.x = Bls[ka * SB + colL1];
                    b1.y = Bls[(ka + 1) * SB + colL1];
            acc0 = __builtin_amdgcn_wmma_f32_16x16x4_f32(false, a, false, b0,
                                                         (short)0, acc0, false, false);
            acc1 = __builtin_amdgcn_wmma_f32_16x16x4_f32(false, a, false, b1,
                                                         (short)0, acc1, false, false);
        }
    }

    const int outRow0 = rowBase + mi * 16 + half * 8;
    const int oc0 = colBase + colL0, oc1 = colBase + colL1;
    const float bv0 = bias ? bias[oc0] : 0.f;
    const float bv1 = bias ? bias[oc1] : 0.f;
#pragma unroll
    for (int r = 0; r < 8; ++r) {
        int orow = outRow0 + r;
        if (orow < M) {
            C[(size_t)orow * N + oc0] = acc0[r] + bv0;
            C[(size_t)orow * N + oc1] = acc1[r] + bv1;
        }
    }
}

// ---------------------------------------------------------------------------
__global__ void fill_f32(float* p, float v, int n) {
    int i = blockIdx.x * blockDim.x + threadIdx.x;
    if (i < n) p[i] = v;
}
__global__ void fill_u32(unsigned* p, unsigned v, int n) {
    int i = blockIdx.x * blockDim.x + threadIdx.x;
    if (i < n) p[i] = v;
}

// el[n,h] = <hpre[n,h,:], al[h,:]> ; er similarly (hpre is pre-bias h)
__global__ void node_logits(const float* __restrict__ hpre,
                            const float* __restrict__ al, const float* __restrict__ ar,
                            float* __restrict__ el, float* __restrict__ er, int NHtot) {
    int i = blockIdx.x * blockDim.x + threadIdx.x;
    if (i >= NHtot) return;
    int n = i >> 2, h = i & 3;
    const float* hp  = hpre + (size_t)n * HD + h * FD;
    const float* alh = al + h * FD;
    const float* arh = ar + h * FD;
    float sl = 0.f, sr = 0.f;
#pragma unroll 4
    for (int d = 0; d < FD; ++d) { float v = hp[d]; sl += v * alh[d]; sr += v * arh[d]; }
    el[i] = sl; er[i] = sr;
}

// segment-max of leaky(el[src]+er[dst]) over dst
__global__ void edge_max(const int* __restrict__ src, const int* __restrict__ dst,
                         const float* __restrict__ el, const float* __restrict__ er,
                         unsigned* __restrict__ mU, int EHtot) {
    int i = blockIdx.x * blockDim.x + threadIdx.x;
    if (i >= EHtot) return;
    int e = i >> 2, h = i & 3;
    int s = src[e], t = dst[e];
    float v = leaky(el[s * HH + h] + er[t * HH + h]);
    atomicMax(&mU[t * HH + h], encf(v));
}

// decode m; nodes with no in-edges (still -inf) -> 0  (in place)
__global__ void m_fixup(unsigned* __restrict__ mU, float* __restrict__ mF, int NHtot) {
    int i = blockIdx.x * blockDim.x + threadIdx.x;
    if (i >= NHtot) return;
    float f = decf(mU[i]);
    if (!isfinite(f)) f = 0.f;
    mF[i] = f;
}

// ex = exp(e - m[dst]); denom[dst] += ex
__global__ void edge_exp(const int* __restrict__ src, const int* __restrict__ dst,
                         const float* __restrict__ el, const float* __restrict__ er,
                         const float* __restrict__ mF, float* __restrict__ ex,
                         float* __restrict__ denom, int EHtot) {
    int i = blockIdx.x * blockDim.x + threadIdx.x;
    if (i >= EHtot) return;
    int e = i >> 2, h = i & 3;
    int s = src[e], t = dst[e];
    float v = leaky(el[s * HH + h] + er[t * HH + h]);
    float x = expf(v - mF[t * HH + h]);
    ex[i] = x;
    atomicAdd(&denom[t * HH + h], x);
}

// out[dst] += a * hpre[src] ; a = ex / max(denom[dst],1e-9). One block per edge.
__global__ __launch_bounds__(128)
void edge_aggregate(const int* __restrict__ src, const int* __restrict__ dst,
                    const float* __restrict__ hpre, const float* __restrict__ ex,
                    const float* __restrict__ denom, float* __restrict__ outagg,
                    float* __restrict__ aw) {
    int e = blockIdx.x;
    int d = threadIdx.x;                  // 0..127
    int s = src[e], t = dst[e];
#pragma unroll
    for (int h = 0; h < HH; ++h) {
        float a = ex[e * HH + h] / fmaxf(denom[t * HH + h], 1e-9f);
        if (aw && d == 0) aw[e * HH + h] = a;
        atomicAdd(&outagg[(size_t)t * HD + h * FD + d],
                  a * hpre[(size_t)s * HD + h * FD + d]);
    }
}

// h1 = leaky(LN(mean_h(agg + b1)))  -- one block (128 threads) per node
__global__ __launch_bounds__(128)
void head_mean_ln(const float* __restrict__ agg, const float* __restrict__ b1,
                  const float* __restrict__ g, const float* __restrict__ b,
                  float* __restrict__ out) {
    int n = blockIdx.x, d = threadIdx.x;
    float v = 0.f;
#pragma unroll
    for (int h = 0; h < HH; ++h)
        v += agg[(size_t)n * HD + h * FD + d] + b1[h * FD + d];
    v *= (1.f / HH);
    __shared__ float red[FD];
    red[d] = v; __syncthreads();
    for (int s = 64; s > 0; s >>= 1) { if (d < s) red[d] += red[d + s]; __syncthreads(); }
    float mean = red[0] * (1.f / FD);
    __syncthreads();
    float c = v - mean;
    red[d] = c * c; __syncthreads();
    for (int s = 64; s > 0; s >>= 1) { if (d < s) red[d] += red[d + s]; __syncthreads(); }
    float inv = rsqrtf(red[0] * (1.f / FD) + 1e-5f);
    out[(size_t)n * FD + d] = leaky(c * inv * g[d] + b[d]);
}

// h2 = leaky(agg + b2); per-graph max pool via encoded atomicMax
__global__ void h2_pool(const float* __restrict__ agg, const float* __restrict__ b2,
                        const int* __restrict__ gid, unsigned* __restrict__ pooledU,
                        int total) {
    int i = blockIdx.x * blockDim.x + threadIdx.x;
    if (i >= total) return;
    int n = i >> 9, j = i & 511;
    float v = leaky(agg[i] + b2[j]);
    atomicMax(&pooledU[gid[n] * HD + j], encf(v));
}

__global__ void pool_decode(unsigned* __restrict__ pu, float* __restrict__ pf, int total) {
    int i = blockIdx.x * blockDim.x + threadIdx.x;
    if (i >= total) return;
    float f = decf(pu[i]);
    if (!isfinite(f)) f = 0.f;
    pf[i] = f;
}

// in-place LN over 768 + leaky; one block (256 threads, 3 elems each) per row
__global__ __launch_bounds__(256)
void ln2_leaky(float* __restrict__ z, const float* __restrict__ g,
               const float* __restrict__ b) {
    int row = blockIdx.x, t = threadIdx.x;
    float v[3]; float s = 0.f;
#pragma unroll
    for (int i = 0; i < 3; ++i) { v[i] = z[(size_t)row * FFF + t + i * 256]; s += v[i]; }
    __shared__ float red[256];
    red[t] = s; __syncthreads();
    for (int st = 128; st > 0; st >>= 1) { if (t < st) red[t] += red[t + st]; __syncthreads(); }
    float mean = red[0] * (1.f / FFF);
    __syncthreads();
    s = 0.f;
#pragma unroll
    for (int i = 0; i < 3; ++i) { v[i] -= mean; s += v[i] * v[i]; }
    red[t] = s; __syncthreads();
    for (int st = 128; st > 0; st >>= 1) { if (t < st) red[t] += red[t + st]; __syncthreads(); }
    float inv = rsqrtf(red[0] * (1.f / FFF) + 1e-5f);
#pragma unroll
    for (int i = 0; i < 3; ++i) {
        int j = t + i * 256;
        z[(size_t)row * FFF + j] = leaky(v[i] * inv * g[j] + b[j]);
    }
}

// probs = softmax(z @ Wc + bc); one wave per row
__global__ __launch_bounds__(32)
void logits_softmax(const float* __restrict__ z, const float* __restrict__ Wc,
                    const float* __restrict__ bc, float* __restrict__ probs) {
    int row = blockIdx.x, lane = threadIdx.x;
    float s0 = 0.f, s1 = 0.f;
    for (int k = lane; k < FFF; k += 32) {
        float zv = z[(size_t)row * FFF + k];
        s0 += zv * Wc[k * CC + 0];
        s1 += zv * Wc[k * CC + 1];
    }
    for (int off = 16; off > 0; off >>= 1) {
        s0 += __shfl_down(s0, off);
        s1 += __shfl_down(s1, off);
    }
    if (lane == 0) {
        s0 += bc[0]; s1 += bc[1];
        float mx = fmaxf(s0, s1);
        float e0 = expf(s0 - mx), e1 = expf(s1 - mx);
        float inv = 1.f / (e0 + e1);
        probs[row * CC + 0] = e0 * inv;
        probs[row * CC + 1] = e1 * inv;
    }
}

// ---------------------------------------------------------------------------
static inline int cdiv(int a, int b) { return (a + b - 1) / b; }

extern "C" void kernel_launch(void* const* d_in, const int* in_sizes, int n_in,
                              void* d_out, int out_size, void* d_ws, size_t ws_size,
                              hipStream_t stream) {
    const float* x    = (const float*)d_in[0];
    const int*   src  = (const int*)  d_in[1];
    const int*   dst  = (const int*)  d_in[2];
    const int*   gid  = (const int*)  d_in[3];
    const float* W1   = (const float*)d_in[4];
    const float* al1  = (const float*)d_in[5];
    const float* ar1  = (const float*)d_in[6];
    const float* b1   = (const float*)d_in[7];
    const float* ln_g = (const float*)d_in[8];
    const float* ln_b = (const float*)d_in[9];
    const float* W2   = (const float*)d_in[10];
    const float* al2  = (const float*)d_in[11];
    const float* ar2  = (const float*)d_in[12];
    const float* b2   = (const float*)d_in[13];
    const float* Wa   = (const float*)d_in[14];
    const float* ba   = (const float*)d_in[15];
    const float* Wb   = (const float*)d_in[16];
    const float* bb   = (const float*)d_in[17];
    const float* l2g  = (const float*)d_in[18];
    const float* l2b  = (const float*)d_in[19];
    const float* Wc   = (const float*)d_in[20];
    const float* bc   = (const float*)d_in[21];

    float* probs = (float*)d_out;            // [64,2]
    float* aw    = (float*)d_out + BB * CC;  // [E,H,1]

    // workspace layout (floats)
    float* w      = (float*)d_ws;
    float* bufA   = w;                 w += (size_t)NN * HD;   // h_pre (layer1/2)
    float* bufB   = w;                 w += (size_t)NN * HD;   // aggregated
    float* h1n    = w;                 w += (size_t)NN * FD;   // normalized h1
    float* el     = w;                 w += NN * HH;
    float* er     = w;                 w += NN * HH;
    float* mbuf   = w;                 w += NN * HH;           // u32 enc then f32
    float* denom  = w;                 w += NN * HH;
    float* exbuf  = w;                 w += (size_t)EE * HH;
    float* pooled = w;                 w += BB * HD;           // u32 enc then f32
    float* za     = w;                 w += BB * FD;
    float* zb     = w;                 w += BB * FFF;

    const int NH = NN * HH, EH = EE * HH;

    for (int layer = 0; layer < 2; ++layer) {
        const float* Ain = layer == 0 ? x : h1n;
        const float* al  = layer == 0 ? al1 : al2;
        const float* ar  = layer == 0 ? ar1 : ar2;

        // h_pre = Ain @ W   [10000,512]  (WMMA fp32, async-LDS staged)
        wmma_gemm_tiled<<<dim3(cdiv(NN, BM), HD / BN), 256, GEMM_LDS_BYTES, stream>>>(
            Ain, layer == 0 ? W1 : W2, nullptr, bufA, NN, HD, FD);

        node_logits<<<cdiv(NH, 256), 256, 0, stream>>>(bufA, al, ar, el, er, NH);

        fill_u32<<<cdiv(NH, 256), 256, 0, stream>>>((unsigned*)mbuf, ENC_NEG_INF, NH);
        fill_f32<<<cdiv(NH, 256), 256, 0, stream>>>(denom, 0.f, NH);
        edge_max<<<cdiv(EH, 256), 256, 0, stream>>>(src, dst, el, er, (unsigned*)mbuf, EH);
        m_fixup<<<cdiv(NH, 256), 256, 0, stream>>>((unsigned*)mbuf, mbuf, NH);
        edge_exp<<<cdiv(EH, 256), 256, 0, stream>>>(src, dst, el, er, mbuf, exbuf, denom, EH);

        fill_f32<<<cdiv(NN * HD, 256), 256, 0, stream>>>(bufB, 0.f, NN * HD);
        edge_aggregate<<<EE, 128, 0, stream>>>(src, dst, bufA, exbuf, denom, bufB,
                                               layer == 0 ? nullptr : aw);
        if (layer == 0)
            head_mean_ln<<<NN, 128, 0, stream>>>(bufB, b1, ln_g, ln_b, h1n);
    }

    // per-graph max pooling of leaky(h2 + b2)
    fill_u32<<<cdiv(BB * HD, 256), 256, 0, stream>>>((unsigned*)pooled, ENC_NEG_INF, BB * HD);
    h2_pool<<<cdiv(NN * HD, 256), 256, 0, stream>>>(bufB, b2, gid, (unsigned*)pooled, NN * HD);
    pool_decode<<<cdiv(BB * HD, 256), 256, 0, stream>>>((unsigned*)pooled, pooled, BB * HD);

    // MLP head (WMMA fp32 with fused bias)
    wmma_gemm_tiled<<<dim3(1, FD / BN), 256, GEMM_LDS_BYTES, stream>>>(
        pooled, Wa, ba, za, BB, FD, HD);
    wmma_gemm_tiled<<<dim3(1, FFF / BN), 256, GEMM_LDS_BYTES, stream>>>(
        za, Wb, bb, zb, BB, FFF, FD);
    ln2_leaky<<<BB, 256, 0, stream>>>(zb, l2g, l2b);
    logits_softmax<<<BB, 32, 0, stream>>>(zb, Wc, bc, probs);
}